// Fusion_detr_53996328845525
// MI455X (gfx1250) — compile-verified
//
#include <hip/hip_runtime.h>
#include <math.h>

// ---------------- problem constants (match reference) ----------------
constexpr int Dm   = 256;   // d_model
constexpr int NHh  = 8;     // heads
constexpr int HDd  = 32;    // head dim (== wave32 width)
constexpr int NPT  = 4;     // dec_n_points
constexpr int NLAY = 6;
constexpr int FFd  = 1024;
constexpr int BSb  = 4;
constexpr int HSs  = 96;
constexpr int WSs  = 96;
constexpr int Nn   = HSs * WSs;     // 9216 source tokens per batch
constexpr int NQq  = 300;           // queries
constexpr int TOK  = BSb * NQq;     // 1200 query rows
constexpr int SRCTOK = BSb * Nn;    // 36864 source rows
constexpr float EPSf = 1e-5f;

typedef __attribute__((ext_vector_type(2))) float v2f;
typedef __attribute__((ext_vector_type(8))) float v8f;

// =====================================================================
// FP32 WMMA GEMM:  C[M x Nc] = A[M x K](lda) @ W[K x Nc](LDW) + bias
// One wave per 16x64 output tile: per K-step one A fragment (b64 load)
// feeds FOUR V_WMMA_F32_16X16X4_F32 ops.  LDW is a compile-time template
// parameter so B-operand addressing strength-reduces to one pointer
// bump per K-step plus immediate offsets (no per-load v_add_nc_u64).
//
// Register layouts (CDNA5 ISA 7.12.2):
//   A 16x4 f32 : lanes 0-15 rows M=0..15, v[0]=K0,v[1]=K1 ; lanes 16-31 K2,K3
//   B 4x16 f32 : lanes 0-15 N=0..15, v[0]=row K0,v[1]=row K1; lanes16-31 K2,K3
//   C 16x16    : VGPR v, lanes0-15 -> M=v, lanes16-31 -> M=v+8, N=lane%16
// =====================================================================
template <int LDW>
__global__ __launch_bounds__(32)
void gemm_wmma_f32(const float* __restrict__ A, int lda,
                   const float* __restrict__ W,
                   const float* __restrict__ bias,
                   float* __restrict__ C, int ldc,
                   int M, int Nc, int K, int relu,
                   const unsigned char* __restrict__ rowmask)
{
    const int lane = threadIdx.x & 31;
    const int half = lane >> 4;     // 0: K pair {0,1}, 1: K pair {2,3}
    const int lid  = lane & 15;
    const int n0 = blockIdx.x * 64;
    const int m0 = blockIdx.y * 16;

    int mrd = m0 + lid; if (mrd >= M) mrd = M - 1;        // clamped A row
    const float* arow = A + (size_t)mrd * lda + half * 2;

    const int ncl0 = n0 +  0 + lid, ncl1 = n0 + 16 + lid;
    const int ncl2 = n0 + 32 + lid, ncl3 = n0 + 48 + lid;
    // clamped B columns (OOB lanes read col 0; results discarded at store)
    const float* wb0 = W + (size_t)(half * 2) * LDW + ((ncl0 < Nc) ? ncl0 : 0);
    const float* wb1 = W + (size_t)(half * 2) * LDW + ((ncl1 < Nc) ? ncl1 : 0);
    const float* wb2 = W + (size_t)(half * 2) * LDW + ((ncl2 < Nc) ? ncl2 : 0);
    const float* wb3 = W + (size_t)(half * 2) * LDW + ((ncl3 < Nc) ? ncl3 : 0);

    v8f acc0 = {}, acc1 = {}, acc2 = {}, acc3 = {};
    #pragma unroll 4
    for (int k0 = 0; k0 < K; k0 += 4) {
        v2f a;
        a[0] = arow[k0 + 0];
        a[1] = arow[k0 + 1];
        const size_t ko = (size_t)k0 * LDW;     // LDW constexpr -> immediates
        v2f b0, b1, b2, b3;
        b0[0] = wb0[ko]; b0[1] = wb0[ko + LDW];
        b1[0] = wb1[ko]; b1[1] = wb1[ko + LDW];
        b2[0] = wb2[ko]; b2[1] = wb2[ko + LDW];
        b3[0] = wb3[ko]; b3[1] = wb3[ko + LDW];
        acc0 = __builtin_amdgcn_wmma_f32_16x16x4_f32(false, a, false, b0,
                                                     (short)0, acc0, false, false);
        acc1 = __builtin_amdgcn_wmma_f32_16x16x4_f32(false, a, false, b1,
                                                     (short)0, acc1, false, false);
        acc2 = __builtin_amdgcn_wmma_f32_16x16x4_f32(false, a, false, b2,
                                                     (short)0, acc2, false, false);
        acc3 = __builtin_amdgcn_wmma_f32_16x16x4_f32(false, a, false, b3,
                                                     (short)0, acc3, false, false);
    }

    auto store_tile = [&](const v8f& acc, int ncl) {
        if (ncl >= Nc) return;
        const float bv = bias ? bias[ncl] : 0.0f;
        #pragma unroll
        for (int v = 0; v < 8; ++v) {
            int m = m0 + half * 8 + v;
            if (m < M) {
                float x = acc[v] + bv;
                if (relu) x = fmaxf(x, 0.0f);
                if (rowmask && rowmask[m]) x = 0.0f;
                C[(size_t)m * ldc + ncl] = x;
            }
        }
    };
    store_tile(acc0, ncl0);
    store_tile(acc1, ncl1);
    store_tile(acc2, ncl2);
    store_tile(acc3, ncl3);
}

// ---------------- gating: column sum over N (padded to 16 rows) -------
__global__ __launch_bounds__(256)
void colsum_k(const float* __restrict__ X, float* __restrict__ S)
{
    const int b = blockIdx.x;      // 0..15 (rows >= BS zero-padded)
    const int d = threadIdx.x;
    float s = 0.0f;
    if (b < BSb) {
        const float* p = X + ((size_t)b * Nn) * Dm + d;
        for (int n = 0; n < Nn; ++n) s += p[(size_t)n * Dm];
    }
    S[(size_t)b * Dm + d] = s;
}

// u[b] = W1 * t[b],  c[b] = b1 . t[b],  with t[b] = S2[b] + N*b2
__global__ __launch_bounds__(256)
void make_u_k(const float* __restrict__ W1, const float* __restrict__ b1,
              const float* __restrict__ S2, const float* __restrict__ b2,
              float* __restrict__ U, float* __restrict__ Cv)
{
    const int b = blockIdx.x;   // 0..BS-1
    const int i = threadIdx.x;  // 0..255
    __shared__ float t[Dm];
    __shared__ float red[Dm];
    t[i] = S2[(size_t)b * Dm + i] + (float)Nn * b2[i];
    __syncthreads();
    float u = 0.0f;
    const float* w = W1 + (size_t)i * Dm;   // row i of (D,D) weight
    for (int o = 0; o < Dm; ++o) u += w[o] * t[o];
    U[(size_t)b * Dm + i] = u;
    red[i] = b1[i] * t[i];
    __syncthreads();
    for (int s = 128; s; s >>= 1) { if (i < s) red[i] += red[i + s]; __syncthreads(); }
    if (i == 0) Cv[b] = red[0];
}

// gate logit per source token: (x . u[b] + c[b]) / sqrt(D)
__global__ __launch_bounds__(256)
void gate_logit_k(const float* __restrict__ X, const float* __restrict__ U,
                  const float* __restrict__ Cv, float* __restrict__ G)
{
    const int t = blockIdx.x * 256 + threadIdx.x;
    if (t >= SRCTOK) return;
    const int b = t / Nn;
    const float* x = X + (size_t)t * Dm;
    const float* u = U + (size_t)b * Dm;
    float s = 0.0f;
    for (int d = 0; d < Dm; ++d) s += x[d] * u[d];
    G[t] = (s + Cv[b]) * 0.0625f;   // 1/sqrt(256)
}

// src = softmax2(ga,gb) mix of aps/dvs
__global__ __launch_bounds__(256)
void fuse_k(const float* __restrict__ aps, const float* __restrict__ dvs,
            const float* __restrict__ GA, const float* __restrict__ GB,
            float* __restrict__ src)
{
    const size_t i = (size_t)blockIdx.x * 256 + threadIdx.x;
    if (i >= (size_t)SRCTOK * Dm) return;
    const size_t t = i >> 8;
    const float ga = GA[t], gb = GB[t];
    const float m = fmaxf(ga, gb);
    const float ea = expf(ga - m), eb = expf(gb - m);
    const float inv = 1.0f / (ea + eb);
    src[i] = aps[i] * (ea * inv) + dvs[i] * (eb * inv);
}

// broadcast query embeds -> qpos / out
__global__ __launch_bounds__(256)
void init_q_k(const float* __restrict__ qe, float* __restrict__ qpos,
              float* __restrict__ out)
{
    const int row = blockIdx.x;       // b*NQ + q
    const int q = row % NQq;
    const int d = threadIdx.x;
    qpos[(size_t)row * Dm + d] = qe[(size_t)q * 2 * Dm + d];
    out [(size_t)row * Dm + d] = qe[(size_t)q * 2 * Dm + Dm + d];
}

__global__ __launch_bounds__(256)
void sigmoid_k(const float* __restrict__ in, float* __restrict__ out, int n)
{
    const int i = blockIdx.x * 256 + threadIdx.x;
    if (i < n) out[i] = 1.0f / (1.0f + expf(-in[i]));
}

__global__ __launch_bounds__(256)
void add_k(const float* __restrict__ a, const float* __restrict__ b,
           float* __restrict__ c, int n)
{
    const int i = blockIdx.x * 256 + threadIdx.x;
    if (i < n) c[i] = a[i] + b[i];
}

// out_row = LayerNorm(out_row + delta_row) * w + b   (row = 256 elems)
__global__ __launch_bounds__(256)
void add_ln_k(float* __restrict__ out, const float* __restrict__ delta,
              const float* __restrict__ w, const float* __restrict__ b)
{
    __shared__ float red[Dm];
    const int row = blockIdx.x;
    const int d = threadIdx.x;
    const size_t i = (size_t)row * Dm + d;
    const float x = out[i] + delta[i];
    red[d] = x; __syncthreads();
    for (int s = 128; s; s >>= 1) { if (d < s) red[d] += red[d + s]; __syncthreads(); }
    const float mu = red[0] * (1.0f / Dm);
    __syncthreads();
    const float c = x - mu;
    red[d] = c * c; __syncthreads();
    for (int s = 128; s; s >>= 1) { if (d < s) red[d] += red[d + s]; __syncthreads(); }
    const float var = red[0] * (1.0f / Dm);
    out[i] = c * rsqrtf(var + EPSf) * w[d] + b[d];
}

// ------- fused self-attention: one wave per (b,h,q), lane = head dim ---
__global__ __launch_bounds__(128)
void self_attn_k(const float* __restrict__ QKP,   // (TOK, 512) q|k
                 const float* __restrict__ VH,    // (TOK, 256)
                 float* __restrict__ HEAD)        // (TOK, 256)
{
    __shared__ float sc[4][NQq];
    __shared__ float qsh[4][32];
    const int wslot = threadIdx.x >> 5;
    const int lane  = threadIdx.x & 31;
    const int w = blockIdx.x * 4 + wslot;          // 0 .. BS*NH*NQ-1 exactly
    const int b = w / (NHh * NQq);
    const int r = w % (NHh * NQq);
    const int h = r / NQq;
    const int q = r % NQq;
    const float inv_sqrt_hd = 0.1767766953f;       // 1/sqrt(32)

    qsh[wslot][lane] = QKP[((size_t)(b * NQq + q)) * 512 + h * 32 + lane];
    __syncthreads();

    // scores: each lane owns keys k = lane, lane+32, ...
    for (int k = lane; k < NQq; k += 32) {
        const float* kv = QKP + ((size_t)(b * NQq + k)) * 512 + Dm + h * 32;
        float s = 0.0f;
        #pragma unroll
        for (int d = 0; d < 32; ++d) s += qsh[wslot][d] * kv[d];
        sc[wslot][k] = s * inv_sqrt_hd;
    }
    // softmax (max/sum via wave reduction)
    float mx = -1e30f;
    for (int k = lane; k < NQq; k += 32) mx = fmaxf(mx, sc[wslot][k]);
    for (int o = 16; o; o >>= 1) mx = fmaxf(mx, __shfl_xor(mx, o, 32));
    float sum = 0.0f;
    for (int k = lane; k < NQq; k += 32) {
        const float e = expf(sc[wslot][k] - mx);
        sc[wslot][k] = e;
        sum += e;
    }
    for (int o = 16; o; o >>= 1) sum += __shfl_xor(sum, o, 32);
    __syncthreads();
    const float rs = 1.0f / sum;

    // weighted sum of V: lane = head dim
    float acc = 0.0f;
    const float* vbase = VH + (size_t)b * NQq * Dm + h * 32 + lane;
    for (int k = 0; k < NQq; ++k)
        acc += sc[wslot][k] * vbase[(size_t)k * Dm];
    HEAD[((size_t)(b * NQq + q)) * Dm + h * 32 + lane] = acc * rs;
}

// ------- MSDeformAttn sampling: one wave per (b,h,q), lane = head dim --
__global__ __launch_bounds__(128)
void ms_deform_k(const float* __restrict__ VALUE,  // (b, N, 256)
                 const float* __restrict__ OFFS,   // (TOK, 64)
                 const float* __restrict__ AWL,    // (TOK, 32)
                 const float* __restrict__ REFP,   // (TOK, 2)
                 const float* __restrict__ VR,     // (BS, 2)
                 float* __restrict__ HEAD)         // (TOK, 256)
{
    const int wslot = threadIdx.x >> 5;
    const int lane  = threadIdx.x & 31;
    const int w = blockIdx.x * 4 + wslot;
    const int b = w / (NHh * NQq);
    const int r = w % (NHh * NQq);
    const int h = r / NQq;
    const int q = r % NQq;
    const size_t tq = (size_t)b * NQq + q;

    // softmax over 4 sampling points (scalar, replicated across lanes)
    float a0 = AWL[tq * 32 + h * 4 + 0];
    float a1 = AWL[tq * 32 + h * 4 + 1];
    float a2 = AWL[tq * 32 + h * 4 + 2];
    float a3 = AWL[tq * 32 + h * 4 + 3];
    const float m = fmaxf(fmaxf(a0, a1), fmaxf(a2, a3));
    a0 = expf(a0 - m); a1 = expf(a1 - m); a2 = expf(a2 - m); a3 = expf(a3 - m);
    const float rsum = 1.0f / (a0 + a1 + a2 + a3);
    const float awp[4] = { a0 * rsum, a1 * rsum, a2 * rsum, a3 * rsum };

    const float rpx = REFP[tq * 2 + 0] * VR[b * 2 + 0];
    const float rpy = REFP[tq * 2 + 1] * VR[b * 2 + 1];

    float acc = 0.0f;
    const float* vbase = VALUE + ((size_t)b * Nn) * Dm + h * 32 + lane;
    #pragma unroll
    for (int p = 0; p < NPT; ++p) {
        const float ox = OFFS[tq * 64 + h * 8 + p * 2 + 0];
        const float oy = OFFS[tq * 64 + h * 8 + p * 2 + 1];
        const float x = (rpx + ox / (float)WSs) * (float)WSs - 0.5f;
        const float y = (rpy + oy / (float)HSs) * (float)HSs - 0.5f;
        const float x0 = floorf(x), y0 = floorf(y);
        const float wx = x - x0, wy = y - y0;
        float s = 0.0f;
        #pragma unroll
        for (int cy = 0; cy < 2; ++cy) {
            #pragma unroll
            for (int cx = 0; cx < 2; ++cx) {
                const float xi = x0 + (float)cx;
                const float yi = y0 + (float)cy;
                const float wgt = (cx ? wx : 1.0f - wx) * (cy ? wy : 1.0f - wy);
                const bool valid = (xi >= 0.0f) && (xi <= (float)(WSs - 1)) &&
                                   (yi >= 0.0f) && (yi <= (float)(HSs - 1));
                int xc = (int)xi; xc = xc < 0 ? 0 : (xc > WSs - 1 ? WSs - 1 : xc);
                int yc = (int)yi; yc = yc < 0 ? 0 : (yc > HSs - 1 ? HSs - 1 : yc);
                const float v = vbase[(size_t)(yc * WSs + xc) * Dm];
                s += valid ? wgt * v : 0.0f;
            }
        }
        acc += awp[p] * s;
    }
    HEAD[tq * Dm + h * 32 + lane] = acc;
}

// final: d_out = [out (TOK*D), ref_pts (TOK*2), ref_pts (TOK*2)]
__global__ __launch_bounds__(256)
void final_copy_k(const float* __restrict__ out, const float* __restrict__ refp,
                  float* __restrict__ dst, int n)
{
    const int i = blockIdx.x * 256 + threadIdx.x;
    if (i >= n) return;
    const int n0 = TOK * Dm;
    const int n1 = n0 + TOK * 2;
    if (i < n0)      dst[i] = out[i];
    else if (i < n1) dst[i] = refp[i - n0];
    else             dst[i] = refp[i - n1];
}

// =====================================================================
extern "C" void kernel_launch(void* const* d_in, const int* in_sizes, int n_in,
                              void* d_out, int out_size, void* d_ws, size_t ws_size,
                              hipStream_t stream)
{
    (void)in_sizes; (void)n_in; (void)ws_size;
    const float* aps    = (const float*)d_in[0];
    const float* dvs    = (const float*)d_in[1];
    const float* qe     = (const float*)d_in[2];
    const float* vr     = (const float*)d_in[3];
    const float* aps_w1 = (const float*)d_in[4];  const float* aps_b1 = (const float*)d_in[5];
    const float* aps_w2 = (const float*)d_in[6];  const float* aps_b2 = (const float*)d_in[7];
    const float* dvs_w1 = (const float*)d_in[8];  const float* dvs_b1 = (const float*)d_in[9];
    const float* dvs_w2 = (const float*)d_in[10]; const float* dvs_b2 = (const float*)d_in[11];
    const float* ref_w  = (const float*)d_in[12]; const float* ref_b  = (const float*)d_in[13];
    const float* qkv_w  = (const float*)d_in[14]; const float* qkv_b  = (const float*)d_in[15];
    const float* saow   = (const float*)d_in[16]; const float* saob   = (const float*)d_in[17];
    const float* off_w  = (const float*)d_in[18]; const float* off_b  = (const float*)d_in[19];
    const float* attn_w = (const float*)d_in[20]; const float* attn_b = (const float*)d_in[21];
    const float* val_w  = (const float*)d_in[22]; const float* val_b  = (const float*)d_in[23];
    const float* cout_w = (const float*)d_in[24]; const float* cout_b = (const float*)d_in[25];
    const float* ln1_w  = (const float*)d_in[26]; const float* ln1_b  = (const float*)d_in[27];
    const float* ln2_w  = (const float*)d_in[28]; const float* ln2_b  = (const float*)d_in[29];
    const float* ln3_w  = (const float*)d_in[30]; const float* ln3_b  = (const float*)d_in[31];
    const float* ffw1   = (const float*)d_in[32]; const float* ffb1   = (const float*)d_in[33];
    const float* ffw2   = (const float*)d_in[34]; const float* ffb2   = (const float*)d_in[35];
    const unsigned char* mask = (const unsigned char*)d_in[36];
    // d_in[37] spatial_shapes: compile-time constants (96,96)

    // -------- workspace carve-up (floats) --------
    float* ws = (float*)d_ws;
    size_t o = 0;
    auto carve = [&](size_t n) { float* p = ws + o; o += n; return p; };
    float* SRC   = carve((size_t)SRCTOK * Dm);   // fused source
    float* VALUE = carve((size_t)SRCTOK * Dm);   // per-layer value proj
    float* OUT   = carve((size_t)TOK * Dm);
    float* QPOS  = carve((size_t)TOK * Dm);
    float* QK    = carve((size_t)TOK * Dm);      // out + qpos
    float* QKP   = carve((size_t)TOK * 512);     // q|k projections
    float* VH    = carve((size_t)TOK * Dm);
    float* HEAD  = carve((size_t)TOK * Dm);      // sa / ca head output
    float* DELTA = carve((size_t)TOK * Dm);
    float* FFH   = carve((size_t)TOK * FFd);
    float* OFFS  = carve((size_t)TOK * 64);
    float* AWL   = carve((size_t)TOK * 32);
    float* REFL  = carve((size_t)TOK * 2);
    float* REFP  = carve((size_t)TOK * 2);
    float* GA    = carve((size_t)SRCTOK);
    float* GB    = carve((size_t)SRCTOK);
    float* ASUM  = carve((size_t)16 * Dm);       // padded to 16 rows for WMMA
    float* S2    = carve((size_t)16 * Dm);
    float* UVEC  = carve((size_t)BSb * Dm);
    float* CVEC  = carve((size_t)16);

    // dispatch with compile-time LDW (weight row stride)
    auto gemm = [&](const float* A, int lda, const float* W, int ldw,
                    const float* bias, float* C, int ldc,
                    int M, int Nc, int K, int relu, const unsigned char* rm) {
        dim3 g((Nc + 63) / 64, (M + 15) / 16);
        dim3 blk(32);
        switch (ldw) {
        case 2:
            gemm_wmma_f32<2><<<g, blk, 0, stream>>>(A, lda, W, bias, C, ldc, M, Nc, K, relu, rm); break;
        case 32:
            gemm_wmma_f32<32><<<g, blk, 0, stream>>>(A, lda, W, bias, C, ldc, M, Nc, K, relu, rm); break;
        case 64:
            gemm_wmma_f32<64><<<g, blk, 0, stream>>>(A, lda, W, bias, C, ldc, M, Nc, K, relu, rm); break;
        case 256:
            gemm_wmma_f32<256><<<g, blk, 0, stream>>>(A, lda, W, bias, C, ldc, M, Nc, K, relu, rm); break;
        case 768:
            gemm_wmma_f32<768><<<g, blk, 0, stream>>>(A, lda, W, bias, C, ldc, M, Nc, K, relu, rm); break;
        case 1024:
            gemm_wmma_f32<1024><<<g, blk, 0, stream>>>(A, lda, W, bias, C, ldc, M, Nc, K, relu, rm); break;
        default: // unreachable for this model
            gemm_wmma_f32<256><<<g, blk, 0, stream>>>(A, lda, W, bias, C, ldc, M, Nc, K, relu, rm); break;
        }
    };

    // ---------------- cross-modal gating + fusion ----------------
    // t[b] = (sum_n aps[b]) @ W2 + N*b2 ; gate = (aps . (W1 t) + b1.t)/sqrt(D)
    colsum_k<<<16, 256, 0, stream>>>(aps, ASUM);
    gemm(ASUM, Dm, aps_w2, Dm, nullptr, S2, Dm, 16, Dm, Dm, 0, nullptr);
    make_u_k<<<BSb, 256, 0, stream>>>(aps_w1, aps_b1, S2, aps_b2, UVEC, CVEC);
    gate_logit_k<<<(SRCTOK + 255) / 256, 256, 0, stream>>>(aps, UVEC, CVEC, GA);

    colsum_k<<<16, 256, 0, stream>>>(dvs, ASUM);
    gemm(ASUM, Dm, dvs_w2, Dm, nullptr, S2, Dm, 16, Dm, Dm, 0, nullptr);
    make_u_k<<<BSb, 256, 0, stream>>>(dvs_w1, dvs_b1, S2, dvs_b2, UVEC, CVEC);
    gate_logit_k<<<(SRCTOK + 255) / 256, 256, 0, stream>>>(dvs, UVEC, CVEC, GB);

    fuse_k<<<(int)(((size_t)SRCTOK * Dm + 255) / 256), 256, 0, stream>>>(
        aps, dvs, GA, GB, SRC);

    // ---------------- query init + reference points ----------------
    init_q_k<<<TOK, 256, 0, stream>>>(qe, QPOS, OUT);
    gemm(QPOS, Dm, ref_w, 2, ref_b, REFL, 2, TOK, 2, Dm, 0, nullptr);
    sigmoid_k<<<(TOK * 2 + 255) / 256, 256, 0, stream>>>(REFL, REFP, TOK * 2);

    const int nTok = TOK * Dm;
    const int saBlocks = (BSb * NHh * NQq) / 4;   // 2400, exact

    // ---------------- decoder layers ----------------
    for (int l = 0; l < NLAY; ++l) {
        const float* qkvWl = qkv_w + (size_t)l * Dm * 3 * Dm;
        const float* qkvBl = qkv_b + (size_t)l * 3 * Dm;

        // --- self attention ---
        add_k<<<(nTok + 255) / 256, 256, 0, stream>>>(OUT, QPOS, QK, nTok);
        gemm(QK,  Dm, qkvWl,            3 * Dm, qkvBl,          QKP, 512, TOK, 512, Dm, 0, nullptr);
        gemm(OUT, Dm, qkvWl + 2 * Dm,   3 * Dm, qkvBl + 2 * Dm, VH,  Dm,  TOK, Dm,  Dm, 0, nullptr);
        self_attn_k<<<saBlocks, 128, 0, stream>>>(QKP, VH, HEAD);
        gemm(HEAD, Dm, saow + (size_t)l * Dm * Dm, Dm, saob + (size_t)l * Dm,
             DELTA, Dm, TOK, Dm, Dm, 0, nullptr);
        add_ln_k<<<TOK, 256, 0, stream>>>(OUT, DELTA,
             ln2_w + (size_t)l * Dm, ln2_b + (size_t)l * Dm);

        // --- deformable cross attention ---
        add_k<<<(nTok + 255) / 256, 256, 0, stream>>>(OUT, QPOS, QK, nTok);
        gemm(SRC, Dm, val_w + (size_t)l * Dm * Dm, Dm, val_b + (size_t)l * Dm,
             VALUE, Dm, SRCTOK, Dm, Dm, 0, mask);
        gemm(QK, Dm, off_w  + (size_t)l * Dm * 64, 64, off_b  + (size_t)l * 64,
             OFFS, 64, TOK, 64, Dm, 0, nullptr);
        gemm(QK, Dm, attn_w + (size_t)l * Dm * 32, 32, attn_b + (size_t)l * 32,
             AWL, 32, TOK, 32, Dm, 0, nullptr);
        ms_deform_k<<<saBlocks, 128, 0, stream>>>(VALUE, OFFS, AWL, REFP, vr, HEAD);
        gemm(HEAD, Dm, cout_w + (size_t)l * Dm * Dm, Dm, cout_b + (size_t)l * Dm,
             DELTA, Dm, TOK, Dm, Dm, 0, nullptr);
        add_ln_k<<<TOK, 256, 0, stream>>>(OUT, DELTA,
             ln1_w + (size_t)l * Dm, ln1_b + (size_t)l * Dm);

        // --- FFN ---
        gemm(OUT, Dm, ffw1 + (size_t)l * Dm * FFd, FFd, ffb1 + (size_t)l * FFd,
             FFH, FFd, TOK, FFd, Dm, /*relu=*/1, nullptr);
        gemm(FFH, FFd, ffw2 + (size_t)l * FFd * Dm, Dm, ffb2 + (size_t)l * Dm,
             DELTA, Dm, TOK, Dm, FFd, 0, nullptr);
        add_ln_k<<<TOK, 256, 0, stream>>>(OUT, DELTA,
             ln3_w + (size_t)l * Dm, ln3_b + (size_t)l * Dm);
    }

    // ---------------- pack outputs: (out, ref_pts, ref_pts) ----------
    final_copy_k<<<(out_size + 255) / 256, 256, 0, stream>>>(
        OUT, REFP, (float*)d_out, out_size);
}